// Attention_55130200211743
// MI455X (gfx1250) — compile-verified
//
#include <hip/hip_runtime.h>
#include <cstdint>

typedef float v2f __attribute__((ext_vector_type(2)));
typedef float v4f __attribute__((ext_vector_type(4)));
typedef float v8f __attribute__((ext_vector_type(8)));

#define B_DIM   4096
#define S_DIM   200
#define KD      128
#define QD      128
#define S_TILES 13                      // ceil(200/16)
#define ROW_P   132                     // padded LDS row stride (floats); 132%64==4 -> conflict-free b64 frags
#define NEG_INF_F (-4294967296.0f)      // float(-2^32+1) rounds to -2^32 in fp32

// ---------------------------------------------------------------------------
// Kernel 1: qp[b,k] = sum_q query[b,q] * w[k,q]   (GEMM M=4096, N=128, K=128)
// grid.x = 4096/16 = 256 blocks, 256 threads (8 waves); wave n owns N-tile n.
// ---------------------------------------------------------------------------
__global__ __launch_bounds__(256) void qproj_wmma(const float* __restrict__ query,
                                                  const float* __restrict__ w,
                                                  float* __restrict__ qp) {
  const int lane = threadIdx.x & 31;
  const int wave = threadIdx.x >> 5;
  const int m0   = blockIdx.x * 16;
  const int n0   = wave * 16;
  const int row  = lane & 15;
  const int hi   = (lane >> 4) & 1;     // lanes 16-31 hold K+2 / K+3

  const float* aBase = query + (size_t)(m0 + row) * QD;  // A[m][q] = query[m0+m, q]
  const float* bBase = w     + (size_t)(n0 + row) * QD;  // B[q][n] = w[n0+n, q]

  v8f c = {};
#pragma unroll
  for (int j = 0; j < QD / 4; ++j) {
    const int q0 = j * 4 + hi * 2;
    v2f a = *(const v2f*)(aBase + q0);
    v2f b = *(const v2f*)(bBase + q0);
    c = __builtin_amdgcn_wmma_f32_16x16x4_f32(false, a, false, b,
                                              (short)0, c, false, false);
  }

  const int n = row;
#pragma unroll
  for (int r = 0; r < 8; ++r) {
    const int m = r + hi * 8;
    qp[(size_t)(m0 + m) * KD + n0 + n] = c[r];
  }
}

// ---------------------------------------------------------------------------
// Kernel 2: async-stage keys[b] slab to LDS, WMMA scores, mask, softmax.
// grid.x = 4096 blocks, 256 threads (8 wave32). Wave w owns s-tiles w, w+8.
// ---------------------------------------------------------------------------
__global__ __launch_bounds__(256) void attn_softmax(const float* __restrict__ keys,
                                                    const int* __restrict__ seq_len,
                                                    const float* __restrict__ qp,
                                                    float* __restrict__ out) {
  __shared__ float keys_s[S_DIM * ROW_P];        // 200*132*4 = 105,600 B
  __shared__ float qp_s[KD];
  __shared__ float score_s[S_TILES * 16];
  __shared__ float red[8];

  const int b = blockIdx.x;
  const int t = threadIdx.x;
  const float* kb = keys + (size_t)b * S_DIM * KD;

  // ---- async copy keys[b] (6400 float4) global -> LDS, coalesced 512B/wave-op,
  //      each HBM line consumed exactly once; padded LDS rows (ROW_P) ----
#pragma unroll
  for (int kk = 0; kk < 25; ++kk) {
    const int i   = t + kk * 256;                // float4 index
    const int row = i >> 5;                      // 32 float4 per source row
    const int c4  = i & 31;
    const uint64_t gaddr = (uint64_t)(uintptr_t)(kb + (size_t)i * 4);
    const uint32_t laddr = (uint32_t)(uintptr_t)&keys_s[row * ROW_P + (c4 << 2)];
    asm volatile("global_load_async_to_lds_b128 %0, %1, off"
                 :: "v"(laddr), "v"(gaddr)
                 : "memory");
  }

  if (t < KD) qp_s[t] = qp[(size_t)b * KD + t];
  const int sl = seq_len[b];

  asm volatile("s_wait_asynccnt 0" ::: "memory");
  __syncthreads();

  // ---- per-wave WMMA score tiles (A = keys rows from LDS, B = qp broadcast) ----
  const int lane = t & 31;
  const int wave = t >> 5;
  const int row  = lane & 15;
  const int hi   = (lane >> 4) & 1;

  for (int tile = wave; tile < S_TILES; tile += 8) {   // wave-uniform loop
    const int s0 = tile * 16;
    int srow = s0 + row;
    if (srow > S_DIM - 1) srow = S_DIM - 1;            // clamp OOB rows (masked later)
    const float* aBase = &keys_s[srow * ROW_P];

    v8f c = {};
#pragma unroll
    for (int j = 0; j < KD / 4; ++j) {
      const int q0 = j * 4 + hi * 2;
      v2f a = *(const v2f*)(aBase + q0);               // ds_load_b64, conflict-free
      v2f bb;
      bb.x = qp_s[q0];                                  // LDS broadcast
      bb.y = qp_s[q0 + 1];
      c = __builtin_amdgcn_wmma_f32_16x16x4_f32(false, a, false, bb,
                                                (short)0, c, false, false);
    }

    if (row == 0) {                                    // lanes 0 and 16 hold all rows
#pragma unroll
      for (int r = 0; r < 8; ++r) {
        const int s = s0 + r + hi * 8;
        score_s[s] = (s < sl) ? c[r] : NEG_INF_F;
      }
    }
  }
  __syncthreads();

  // ---- softmax over score_s[0..S_DIM-1] (one element per thread) ----
  const float x = (t < S_DIM) ? score_s[t] : NEG_INF_F;

  float m = x;
#pragma unroll
  for (int off = 16; off > 0; off >>= 1) m = fmaxf(m, __shfl_xor(m, off, 32));
  if (lane == 0) red[wave] = m;
  __syncthreads();
  if (t == 0) {
    float mm = red[0];
    for (int i = 1; i < 8; ++i) mm = fmaxf(mm, red[i]);
    red[0] = mm;
  }
  __syncthreads();
  const float maxv = red[0];

  const float e = (t < S_DIM) ? __expf(x - maxv) : 0.0f;
  float s = e;
#pragma unroll
  for (int off = 16; off > 0; off >>= 1) s += __shfl_xor(s, off, 32);
  __syncthreads();                      // red[] reuse guard
  if (lane == 0) red[wave] = s;
  __syncthreads();
  if (t == 0) {
    float ss = 0.0f;
    for (int i = 0; i < 8; ++i) ss += red[i];
    red[0] = ss;
  }
  __syncthreads();
  const float inv = 1.0f / red[0];

  if (t < S_DIM) out[(size_t)b * S_DIM + t] = e * inv;
}

// ---------------------------------------------------------------------------
extern "C" void kernel_launch(void* const* d_in, const int* in_sizes, int n_in,
                              void* d_out, int out_size, void* d_ws, size_t ws_size,
                              hipStream_t stream) {
  (void)in_sizes; (void)n_in; (void)out_size; (void)ws_size;
  const float* query = (const float*)d_in[0];   // [4096,1,128] fp32
  const float* keys  = (const float*)d_in[1];   // [4096,200,128] fp32
  const int*   seql  = (const int*)d_in[2];     // [4096,1] int
  const float* w     = (const float*)d_in[3];   // [128,128] fp32
  float*       out   = (float*)d_out;           // [4096,200] fp32
  float*       qp    = (float*)d_ws;            // scratch: 4096*128 fp32 = 2 MB

  qproj_wmma  <<<B_DIM / 16, 256, 0, stream>>>(query, w, qp);
  attn_softmax<<<B_DIM,      256, 0, stream>>>(keys, seql, qp, out);
}